// GraphQNetwork_46067819217490
// MI455X (gfx1250) — compile-verified
//
#include <hip/hip_runtime.h>

typedef float v2f __attribute__((ext_vector_type(2)));
typedef float v8f __attribute__((ext_vector_type(8)));

// ---------- atomic helpers: force native FP atomics (no CAS loops) ----------
__device__ __forceinline__ void atomAddG(float* p, float v) {
  __hip_atomic_fetch_add(p, v, __ATOMIC_RELAXED, __HIP_MEMORY_SCOPE_AGENT);
}
__device__ __forceinline__ void atomAddL(float* p, float v) {
  __hip_atomic_fetch_add(p, v, __ATOMIC_RELAXED, __HIP_MEMORY_SCOPE_WORKGROUP);
}

// ---------- elementwise utility kernels ----------
__global__ void k_zero(float* __restrict__ p, long n) {
  long i = (long)blockIdx.x * blockDim.x + threadIdx.x;
  if (i < n) p[i] = 0.0f;
}

// indeg accumulation: deg[dst[e]] += 1
__global__ void k_deg(const int* __restrict__ dst, float* __restrict__ deg, long E) {
  long e = (long)blockIdx.x * blockDim.x + threadIdx.x;
  if (e < E) atomAddG(&deg[dst[e]], 1.0f);
}

// dinv[i] = rsqrt(indeg + 1)   (self-loop included; always > 0)
__global__ void k_dinv(float* __restrict__ deg, long n) {
  long i = (long)blockIdx.x * blockDim.x + threadIdx.x;
  if (i < n) deg[i] = rsqrtf(deg[i] + 1.0f);
}

// acc[i*F+f] = dinv[i]^2 * hw[i*F+f]   (self-loop message; initializes accumulator)
__global__ void k_selfinit(const float* __restrict__ hw, const float* __restrict__ dinv,
                           float* __restrict__ acc, long total, int logF) {
  long t = (long)blockIdx.x * blockDim.x + threadIdx.x;
  if (t >= total) return;
  long i = t >> logF;
  float dv = dinv[i];
  acc[t] = dv * dv * hw[t];
}

// edge scatter: lane = (edge, feature). Coalesced h[src] reads, contiguous FP atomics.
__global__ void k_scatter(const int* __restrict__ src, const int* __restrict__ dst,
                          const float* __restrict__ dinv, const float* __restrict__ h,
                          float* __restrict__ acc, long E, int logF) {
  long t = (long)blockIdx.x * blockDim.x + threadIdx.x;
  long total = E << logF;
  if (t >= total) return;
  long e = t >> logF;
  int  f = (int)(t & ((1 << logF) - 1));
  int  s = src[e], d = dst[e];
  float w = dinv[s] * dinv[d];
  atomAddG(&acc[((long)d << logF) + f], w * h[((long)s << logF) + f]);
}

__global__ void k_biasrelu(const float* __restrict__ acc, const float* __restrict__ b,
                           float* __restrict__ out, long total, int logF) {
  long t = (long)blockIdx.x * blockDim.x + threadIdx.x;
  if (t >= total) return;
  int f = (int)(t & ((1 << logF) - 1));
  float v = acc[t] + b[f];
  out[t] = v > 0.0f ? v : 0.0f;
}

// ---------- WMMA GEMM: hw1 = x[N,128] @ W1[128,16] (fp32, fp32 accumulate) ----------
__global__ void k_gemm1(const float* __restrict__ x, const float* __restrict__ W,
                        float* __restrict__ out, int N) {
  __shared__ float sW[128 * 16];
  for (int i = threadIdx.x; i < 128 * 16; i += blockDim.x) sW[i] = W[i];
  __syncthreads();

  int wavesPerBlock = blockDim.x >> 5;
  int tile = blockIdx.x * wavesPerBlock + (threadIdx.x >> 5);
  int numTiles = (N + 15) >> 4;
  if (tile >= numTiles) return;            // wave-uniform; EXEC stays all-ones

  int lane = threadIdx.x & 31;
  int half = lane >> 4;                    // 0: K+0/K+1, 1: K+2/K+3
  int l    = lane & 15;
  int rowBase = tile << 4;
  int arow = rowBase + l; if (arow > N - 1) arow = N - 1;
  const float* xrow = x + (long)arow * 128;

  v8f acc = {};
  #pragma unroll
  for (int k = 0; k < 128; k += 4) {
    int ka = k + 2 * half;
    v2f a; a.x = xrow[ka];            a.y = xrow[ka + 1];
    v2f b; b.x = sW[ka * 16 + l];     b.y = sW[(ka + 1) * 16 + l];
    acc = __builtin_amdgcn_wmma_f32_16x16x4_f32(false, a, false, b, (short)0, acc,
                                                false, false);
  }
  #pragma unroll
  for (int r = 0; r < 8; r++) {
    int m = rowBase + half * 8 + r;
    if (m < N) out[(long)m * 16 + l] = acc[r];
  }
}

// ---------- WMMA GEMM: hw2 = h1[N,16] @ W2[16,32] ----------
__global__ void k_gemm2(const float* __restrict__ h, const float* __restrict__ W,
                        float* __restrict__ out, int N) {
  __shared__ float sW[16 * 32];
  for (int i = threadIdx.x; i < 16 * 32; i += blockDim.x) sW[i] = W[i];
  __syncthreads();

  int wavesPerBlock = blockDim.x >> 5;
  int tile = blockIdx.x * wavesPerBlock + (threadIdx.x >> 5);
  int numTiles = (N + 15) >> 4;
  if (tile >= numTiles) return;

  int lane = threadIdx.x & 31;
  int half = lane >> 4;
  int l    = lane & 15;
  int rowBase = tile << 4;
  int arow = rowBase + l; if (arow > N - 1) arow = N - 1;
  const float* hrow = h + (long)arow * 16;

  v8f acc0 = {}, acc1 = {};
  #pragma unroll
  for (int k = 0; k < 16; k += 4) {
    int ka = k + 2 * half;
    v2f a;  a.x  = hrow[ka];                a.y  = hrow[ka + 1];
    v2f b0; b0.x = sW[ka * 32 + l];         b0.y = sW[(ka + 1) * 32 + l];
    v2f b1; b1.x = sW[ka * 32 + 16 + l];    b1.y = sW[(ka + 1) * 32 + 16 + l];
    acc0 = __builtin_amdgcn_wmma_f32_16x16x4_f32(false, a, false, b0, (short)0, acc0,
                                                 false, false);
    acc1 = __builtin_amdgcn_wmma_f32_16x16x4_f32(false, a, false, b1, (short)0, acc1,
                                                 false, false);
  }
  #pragma unroll
  for (int r = 0; r < 8; r++) {
    int m = rowBase + half * 8 + r;
    if (m < N) {
      out[(long)m * 32 + l]      = acc0[r];
      out[(long)m * 32 + 16 + l] = acc1[r];
    }
  }
}

// ---------- pooling: LDS-reduced segment sum over sorted batch ----------
#define POOL_NODES 2048
__global__ void k_pool(const float* __restrict__ h2, const int* __restrict__ batch,
                       float* __restrict__ gsum, float* __restrict__ gcnt, int N) {
  __shared__ float ls[64 * 32];
  __shared__ float lc[64];
  for (int i = threadIdx.x; i < 64 * 32; i += blockDim.x) ls[i] = 0.0f;
  for (int i = threadIdx.x; i < 64;      i += blockDim.x) lc[i] = 0.0f;
  __syncthreads();

  long base = (long)blockIdx.x * POOL_NODES;
  long end  = base + POOL_NODES; if (end > N) end = N;
  for (long t = base * 32 + threadIdx.x; t < end * 32; t += blockDim.x) {
    long i = t >> 5;
    int  f = (int)(t & 31);
    int  g = batch[i];
    atomAddL(&ls[g * 32 + f], h2[t]);
    if (f == 0) atomAddL(&lc[g], 1.0f);
  }
  __syncthreads();
  for (int i = threadIdx.x; i < 64 * 32; i += blockDim.x) {
    float v = ls[i]; if (v != 0.0f) atomAddG(&gsum[i], v);
  }
  for (int i = threadIdx.x; i < 64; i += blockDim.x) {
    float v = lc[i]; if (v != 0.0f) atomAddG(&gcnt[i], v);
  }
}

__global__ void k_gdiv(const float* __restrict__ gsum, const float* __restrict__ gcnt,
                       float* __restrict__ g) {
  int t = blockIdx.x * blockDim.x + threadIdx.x;
  if (t >= 64 * 32) return;
  float c = gcnt[t >> 5];
  c = c < 1.0f ? 1.0f : c;
  g[t] = gsum[t] / c;
}

// ---------- tiny MLP head ----------
__global__ void k_fc1(const float* __restrict__ g, const float* __restrict__ W,
                      const float* __restrict__ b, float* __restrict__ out) {
  int t = blockIdx.x * blockDim.x + threadIdx.x;
  if (t >= 64 * 64) return;
  int gi = t >> 6, o = t & 63;
  float s = b[o];
  #pragma unroll
  for (int k = 0; k < 32; k++) s += g[gi * 32 + k] * W[k * 64 + o];
  out[t] = s > 0.0f ? s : 0.0f;
}

__global__ void k_fc2(const float* __restrict__ gh, const float* __restrict__ W,
                      const float* __restrict__ b, float* __restrict__ out) {
  int t = blockIdx.x * blockDim.x + threadIdx.x;
  if (t >= 64 * 8) return;
  int gi = t >> 3, o = t & 7;
  float s = b[o];
  #pragma unroll
  for (int k = 0; k < 64; k++) s += gh[gi * 64 + k] * W[k * 8 + o];
  out[t] = s;
}

// ---------- driver ----------
extern "C" void kernel_launch(void* const* d_in, const int* in_sizes, int n_in,
                              void* d_out, int out_size, void* d_ws, size_t ws_size,
                              hipStream_t stream) {
  const float* x    = (const float*)d_in[0];
  const int*   ei   = (const int*)  d_in[1];
  const int*   bat  = (const int*)  d_in[2];
  const float* W1   = (const float*)d_in[3];
  const float* b1   = (const float*)d_in[4];
  const float* W2   = (const float*)d_in[5];
  const float* b2   = (const float*)d_in[6];
  const float* Wf1  = (const float*)d_in[7];
  const float* bf1  = (const float*)d_in[8];
  const float* Wf2  = (const float*)d_in[9];
  const float* bf2  = (const float*)d_in[10];
  float* out = (float*)d_out;

  long N = in_sizes[0] / 128;
  long E = in_sizes[1] / 2;
  const int* src = ei;
  const int* dst = ei + E;

  float* ws   = (float*)d_ws;
  float* dinv = ws;  ws += N;
  float* hw1  = ws;  ws += N * 16;   // also reused as h1 after bias+relu
  float* acc1 = ws;  ws += N * 16;
  float* hw2  = ws;  ws += N * 32;   // also reused as h2
  float* acc2 = ws;  ws += N * 32;
  float* gsum = ws;  ws += 64 * 32;  // gcnt must follow contiguously (zeroed together)
  float* gcnt = ws;  ws += 64;
  float* g    = ws;  ws += 64 * 32;
  float* gh   = ws;  ws += 64 * 64;

  dim3 blk(256);
  auto nb = [](long n) { return dim3((unsigned)((n + 255) / 256)); };

  // degrees -> dinv
  k_zero<<<nb(N), blk, 0, stream>>>(dinv, N);
  k_deg <<<nb(E), blk, 0, stream>>>(dst, dinv, E);
  k_dinv<<<nb(N), blk, 0, stream>>>(dinv, N);

  long tiles  = (N + 15) / 16;
  dim3 gblocks((unsigned)((tiles + 7) / 8));   // 8 waves / block, 1 tile / wave

  // layer 1: hw1 = x @ W1 ; aggregate ; relu
  k_gemm1   <<<gblocks, blk, 0, stream>>>(x, W1, hw1, (int)N);
  k_selfinit<<<nb(N * 16), blk, 0, stream>>>(hw1, dinv, acc1, N * 16, 4);
  k_scatter <<<nb(E * 16), blk, 0, stream>>>(src, dst, dinv, hw1, acc1, E, 4);
  k_biasrelu<<<nb(N * 16), blk, 0, stream>>>(acc1, b1, hw1, N * 16, 4);

  // layer 2: hw2 = h1 @ W2 ; aggregate ; relu
  k_gemm2   <<<gblocks, blk, 0, stream>>>(hw1, W2, hw2, (int)N);
  k_selfinit<<<nb(N * 32), blk, 0, stream>>>(hw2, dinv, acc2, N * 32, 5);
  k_scatter <<<nb(E * 32), blk, 0, stream>>>(src, dst, dinv, hw2, acc2, E, 5);
  k_biasrelu<<<nb(N * 32), blk, 0, stream>>>(acc2, b2, hw2, N * 32, 5);

  // mean pool + MLP head
  k_zero<<<nb(64 * 32 + 64), blk, 0, stream>>>(gsum, 64 * 32 + 64);
  k_pool<<<dim3((unsigned)((N + POOL_NODES - 1) / POOL_NODES)), blk, 0, stream>>>(
      hw2, bat, gsum, gcnt, (int)N);
  k_gdiv<<<dim3(8),  blk, 0, stream>>>(gsum, gcnt, g);
  k_fc1 <<<dim3(16), blk, 0, stream>>>(g, Wf1, bf1, gh);
  k_fc2 <<<dim3(2),  blk, 0, stream>>>(gh, Wf2, bf2, out);
}